// Pitch_27041114095643
// MI455X (gfx1250) — compile-verified
//
#include <hip/hip_runtime.h>
#include <hip/hip_bf16.h>

typedef __attribute__((ext_vector_type(2))) float v2f;
typedef __attribute__((ext_vector_type(8))) float v8f;

#define BB_   16
#define TT_   88200
#define WW_   2048
#define STEP_ 256
#define FF_   337      // (88200-2048)/256 + 1
#define TAU_  2048
#define KLEN  2064     // 129*16 : K extent of the Toeplitz GEMM
#define NJ    129      // KLEN/16 outer iterations
#define NT    288      // 8+ waves for scans; WMMA phase uses waves 0..2
#define MOUT  80

#define SG(i) ((i) + ((i) >> 4))   // skew used only when FILLING gB (not in hot loop)
#define GA_LEN 2080                // linear copy: A reads idx <= 2078
#define GB_LEN 4656                // skewed copy: max skewed idx 4622

__global__ __launch_bounds__(NT) void pitch_yin_wmma(const float* __restrict__ x,
                                                     float* __restrict__ out)
{
    __shared__ float gA[GA_LEN];     // linear, shifted, zero-padded: gA[i] = f[i-16]
    __shared__ float gB[GB_LEN];     // skewed copy: gB[SG(i)] = f[i-16]
    __shared__ float cs[WW_ + 1];    // prefix sum of squares
    __shared__ float acf[TAU_];      // acf -> later reused to hold cmndfs
    __shared__ float part[256];      // scan partials

    const int tid = threadIdx.x;
    const int fi  = blockIdx.x;          // frame id in [0, B*F)
    const int b   = fi / FF_;
    const int fr  = fi - b * FF_;
    const long base = (long)b * TT_ + (long)fr * STEP_;

    // ---- stage frame into LDS (two copies) with zero padding, shifted by +16 ----
    for (int i = tid; i < GA_LEN; i += NT) gA[i] = 0.0f;
    for (int i = tid; i < GB_LEN; i += NT) gB[i] = 0.0f;
    __syncthreads();
    for (int t = tid; t < WW_; t += NT) {
        const float s = x[base + t];
        gA[16 + t]     = s;
        gB[SG(16 + t)] = s;
    }
    __syncthreads();

    // ---- scan #1: cs[t] = sum_{u<t} f[u]^2 ----
    float v[8];
    if (tid < 256) {
        const int bb = tid * 8;
        float run = 0.f;
        #pragma unroll
        for (int j = 0; j < 8; ++j) { float s = gA[16 + bb + j]; run += s * s; v[j] = run; }
        part[tid] = run;
    }
    __syncthreads();
    for (int off = 1; off < 256; off <<= 1) {
        float add = 0.f;
        if (tid < 256 && tid >= off) add = part[tid - off];
        __syncthreads();
        if (tid < 256) part[tid] += add;
        __syncthreads();
    }
    if (tid < 256) {
        const int bb = tid * 8;
        const float off0 = (tid == 0) ? 0.f : part[tid - 1];
        #pragma unroll
        for (int j = 0; j < 8; ++j) cs[bb + j + 1] = off0 + v[j];
    }
    if (tid == 0) cs[0] = 0.f;
    __syncthreads();

    // ---- WMMA autocorrelation: D[r,q] = sum_k g[k+r]*g[k+16q] = acf(16q - r) ----
    // 3 waves x 3 q-tiles each; skewed B addresses are affine:
    //   SG(16j+4s + 2hi + 16Q) = (2hi + 17Q) + 17j + 4s   (per-lane const + incremental)
    if (tid < 96) {
        const int wave = tid >> 5;    // 0..2
        const int lane = tid & 31;
        const int m    = lane & 15;
        const int hi   = lane >> 4;
        int ia = m + 2 * hi;          // linear A index: gA[kk + m + 2hi]
        int ib0 = 2 * hi + 17 * (((3 * wave + 0) << 4) + m);
        int ib1 = 2 * hi + 17 * (((3 * wave + 1) << 4) + m);
        int ib2 = 2 * hi + 17 * (((3 * wave + 2) << 4) + m);
        v8f c[3];
        c[0] = (v8f){}; c[1] = (v8f){}; c[2] = (v8f){};
        for (int j = 0; j < NJ; ++j) {
            #pragma unroll
            for (int s = 0; s < 4; ++s) {
                v2f a;
                a.x = gA[ia + 4 * s];
                a.y = gA[ia + 4 * s + 1];
                v2f b0v, b1v, b2v;
                b0v.x = gB[ib0 + 4 * s];  b0v.y = gB[ib0 + 4 * s + 1];
                b1v.x = gB[ib1 + 4 * s];  b1v.y = gB[ib1 + 4 * s + 1];
                b2v.x = gB[ib2 + 4 * s];  b2v.y = gB[ib2 + 4 * s + 1];
                c[0] = __builtin_amdgcn_wmma_f32_16x16x4_f32(
                           false, a, false, b0v, (short)0, c[0], false, false);
                c[1] = __builtin_amdgcn_wmma_f32_16x16x4_f32(
                           false, a, false, b1v, (short)0, c[1], false, false);
                c[2] = __builtin_amdgcn_wmma_f32_16x16x4_f32(
                           false, a, false, b2v, (short)0, c[2], false, false);
            }
            ia  += 16;
            ib0 += 17; ib1 += 17; ib2 += 17;
        }
        #pragma unroll
        for (int t = 0; t < 3; ++t) {
            const int q = ((3 * wave + t) << 4) + m;   // D col -> q
            #pragma unroll
            for (int j = 0; j < 8; ++j) {
                const int r   = j + 8 * hi;            // D row -> r
                const int tau = 16 * q - r;
                if (tau >= 0 && tau < TAU_) acf[tau] = c[t][j];
            }
        }
    }
    __syncthreads();

    // ---- df + scan #2 (cumsum of df[1..]) -> cmndfs (reuse acf[]) ----
    float d8[8];
    if (tid < 256) {
        const int bb = tid * 8;
        float run = 0.f;
        #pragma unroll
        for (int j = 0; j < 8; ++j) {
            const int tau = bb + j;
            float d = 0.f;
            if (tau >= 1) d = cs[WW_ - tau] + cs[WW_] - cs[tau] - 2.0f * acf[tau];
            d8[j] = d;
            run += d;
            v[j] = run;           // inclusive cumsum of df starting at tau=1
        }
        part[tid] = run;
    }
    __syncthreads();
    for (int off = 1; off < 256; off <<= 1) {
        float add = 0.f;
        if (tid < 256 && tid >= off) add = part[tid - off];
        __syncthreads();
        if (tid < 256) part[tid] += add;
        __syncthreads();
    }
    if (tid < 256) {
        const int bb = tid * 8;
        const float off0 = (tid == 0) ? 0.f : part[tid - 1];
        #pragma unroll
        for (int j = 0; j < 8; ++j) {
            const int tau = bb + j;
            if (tau == 0) {
                acf[0] = 1.0f;
            } else {
                const float cum = off0 + v[j];
                acf[tau] = d8[j] * (float)tau / (cum + 1e-8f);
            }
        }
    }
    __syncthreads();

    // ---- sample 80 MIDI lags with linear interpolation ----
    if (tid < MOUT) {
        const int midi = 5 + tid;
        const double f    = 440.0 * exp2(((double)midi - 69.0) / 12.0);
        const double cms  = 22050.0 / f;
        const double fl   = floor(cms);
        const double ce   = ceil(cms);
        const float  frac = (float)((cms - fl) / (ce - fl));
        const int fli = (int)fl;
        const int cei = (int)ce;
        const float cf = acf[fli];
        const float y  = (acf[cei] - cf) * frac + cf;
        out[(long)b * (MOUT * FF_) + (long)tid * FF_ + fr] = y;
    }
}

extern "C" void kernel_launch(void* const* d_in, const int* in_sizes, int n_in,
                              void* d_out, int out_size, void* d_ws, size_t ws_size,
                              hipStream_t stream) {
    (void)in_sizes; (void)n_in; (void)d_ws; (void)ws_size; (void)out_size;
    const float* x = (const float*)d_in[0];
    float* out = (float*)d_out;
    pitch_yin_wmma<<<dim3(BB_ * FF_), dim3(NT), 0, stream>>>(x, out);
}